// VariationalQuantumLayer_38895223833036
// MI455X (gfx1250) — compile-verified
//
#include <hip/hip_runtime.h>
#include <math.h>

#define PI_F 3.14159265358979f

typedef float v2f __attribute__((ext_vector_type(2)));
typedef float v4f __attribute__((ext_vector_type(4)));
typedef float v8f __attribute__((ext_vector_type(8)));

// Workspace layout (floats):
//   [0      , 2<<20)  WP  : packed W^T, Re/Im interleaved per K-pair:
//                           WP[j/2][n][4] = { Re(j,n), Re(j+1,n), Im(j,n), Im(j+1,n) }
//   [2<<20  , 3<<20)  lut : 16384 x 64  (lo[32] then hi[32] per batch row)
// Total 12 MB.

// ---------------------------------------------------------------------------
// Kernel 1: build the fixed 1024x1024 circuit operator W (layers only).
// One wave (32 threads) per basis column j; state lives in LDS.
// ---------------------------------------------------------------------------
__global__ __launch_bounds__(32)
void build_operator_kernel(const float* __restrict__ w, float* __restrict__ wp) {
  __shared__ float sre[1024], sim[1024];
  __shared__ float g[30][8];  // fused U = Rz*Ry*Rx per (layer,qubit), row-major 2x2 complex
  const int lane = threadIdx.x;
  const int col  = blockIdx.x;

  if (lane < 30) {
    float a = w[lane * 3 + 0] * 0.5f;
    float b = w[lane * 3 + 1] * 0.5f;
    float c = w[lane * 3 + 2] * 0.5f;
    float cx = cosf(a), sx = sinf(a);
    float cy = cosf(b), sy = sinf(b);
    float cz = cosf(c), sz = sinf(c);
    // M = Ry * Rx
    float m00r =  cy * cx, m00i =  sy * sx;
    float m01r = -sy * cx, m01i = -cy * sx;
    float m10r =  sy * cx, m10i = -cy * sx;
    float m11r =  cy * cx, m11i = -sy * sx;
    // row0 *= e^{-i c};  row1 *= e^{+i c}
    g[lane][0] = m00r * cz + m00i * sz;  g[lane][1] = m00i * cz - m00r * sz;
    g[lane][2] = m01r * cz + m01i * sz;  g[lane][3] = m01i * cz - m01r * sz;
    g[lane][4] = m10r * cz - m10i * sz;  g[lane][5] = m10i * cz + m10r * sz;
    g[lane][6] = m11r * cz - m11i * sz;  g[lane][7] = m11i * cz + m11r * sz;
  }
  for (int i = lane; i < 1024; i += 32) { sre[i] = 0.f; sim[i] = 0.f; }
  __syncthreads();
  if (lane == 0) sre[col] = 1.0f;
  __syncthreads();

  for (int layer = 0; layer < 3; ++layer) {
    // CNOT ring: control i (bit 9-i), target (i+1)%10 (bit 9-((i+1)%10))
    for (int i = 0; i < 10; ++i) {
      const int pc = 9 - i;
      const int pt = 9 - ((i + 1) % 10);
      const int pl = pc < pt ? pc : pt;
      const int ph = pc < pt ? pt : pc;
      for (int t = 0; t < 8; ++t) {
        int e  = t * 32 + lane;  // 8 free bits
        int t1 = ((e  >> pl) << (pl + 1)) | (e  & ((1 << pl) - 1));
        int k0 = ((t1 >> ph) << (ph + 1)) | (t1 & ((1 << ph) - 1));
        int k  = k0 | (1 << pc);        // control=1, target=0
        int kp = k ^ (1 << pt);
        float tr = sre[k], ti = sim[k];
        sre[k]  = sre[kp]; sim[k]  = sim[kp];
        sre[kp] = tr;      sim[kp] = ti;
      }
      __syncthreads();
    }
    // fused 1-qubit gates
    for (int q = 0; q < 10; ++q) {
      const int p = 9 - q;
      const float* u = g[layer * 10 + q];
      const float u00r = u[0], u00i = u[1], u01r = u[2], u01i = u[3];
      const float u10r = u[4], u10i = u[5], u11r = u[6], u11i = u[7];
      for (int t = 0; t < 16; ++t) {
        int pi = t * 32 + lane;
        int k0 = ((pi >> p) << (p + 1)) | (pi & ((1 << p) - 1));
        int k1 = k0 | (1 << p);
        float ar = sre[k0], ai = sim[k0];
        float br = sre[k1], bi = sim[k1];
        sre[k0] = u00r * ar - u00i * ai + u01r * br - u01i * bi;
        sim[k0] = u00r * ai + u00i * ar + u01r * bi + u01i * br;
        sre[k1] = u10r * ar - u10i * ai + u11r * br - u11i * bi;
        sim[k1] = u10r * ai + u10i * ar + u11r * bi + u11i * br;
      }
      __syncthreads();
    }
  }

  // Store W^T row j in Re/Im-interleaved K-pair-packed layout.
  const int jp = col >> 1, par = col & 1;
  float* p = wp + (size_t)jp * 4096 + par;
  for (int n = lane; n < 1024; n += 32) {
    p[n * 4]     = sre[n];
    p[n * 4 + 2] = sim[n];
  }
}

// ---------------------------------------------------------------------------
// Kernel 2: per-batch-row 64-entry product LUT:  psi[b][k] = hi[k>>5]*lo[k&31]
// ---------------------------------------------------------------------------
__global__ __launch_bounds__(256)
void encode_lut_kernel(const float* __restrict__ x, float* __restrict__ lut) {
  const int b = blockIdx.x * 256 + threadIdx.x;
  float cv[10], sv[10];
  #pragma unroll
  for (int q = 0; q < 10; ++q) {
    float h = tanhf(x[(size_t)b * 10 + q]) * (PI_F * 0.5f);
    cv[q] = cosf(h); sv[q] = sinf(h);
  }
  float* o = lut + (size_t)b * 64;
  for (int t = 0; t < 32; ++t) {
    float lo = 1.f, hi = 1.f;
    #pragma unroll
    for (int q = 5; q < 10; ++q) lo *= ((t >> (9 - q)) & 1) ? sv[q] : cv[q];
    #pragma unroll
    for (int q = 0; q < 5;  ++q) hi *= ((t >> (4 - q)) & 1) ? sv[q] : cv[q];
    o[t] = lo; o[32 + t] = hi;
  }
}

// ---------------------------------------------------------------------------
// Kernel 3: Phi = psi * W^T via v_wmma_f32_16x16x4_f32 (Re & Im GEMMs),
// fused |Phi|^2 * Z-sign reduction into out[16384][10].
// Block = 256 threads (8 waves), 16 batch rows x all 1024 states.
// Two passes of 4 N-tiles each (64 acc VGPRs live) so the scheduler has
// register headroom to keep many b128 B-loads in flight.  One b128 load
// yields both the Re and Im B-fragments (even-aligned subpairs).
// ---------------------------------------------------------------------------
__global__ __launch_bounds__(256)
void gemm_expect_kernel(const float* __restrict__ wp,
                        const float* __restrict__ lut, float* __restrict__ out) {
  __shared__ float lut_s[1024];   // 16 rows x 64
  __shared__ float oacc[160];     // 16 rows x 10 wires
  const int tid   = threadIdx.x;
  const int mbase = blockIdx.x * 16;

  for (int i = tid; i < 1024; i += 256) lut_s[i] = lut[(size_t)mbase * 64 + i];
  for (int i = tid; i < 160;  i += 256) oacc[i] = 0.f;
  __syncthreads();

  const int wave  = tid >> 5;
  const int lane  = tid & 31;
  const int row   = lane & 15;   // A-matrix M / B-matrix N lane index
  const int khalf = lane >> 4;   // 0: K={j,j+1}, 1: K={j+2,j+3}
  const int koff  = khalf * 2;
  const float* lutLo = &lut_s[row * 64];
  const float* lutHi = &lut_s[row * 64 + 32];

  for (int half = 0; half < 2; ++half) {
    const int tbase = wave * 8 + half * 4;                 // first N-tile this pass
    const int laneOfs = khalf * 4096 + (tbase * 16 + row) * 4;  // per-lane element ofs

    v8f accR[4] = {{}, {}, {}, {}};
    v8f accI[4] = {{}, {}, {}, {}};

    for (int jc = 0; jc < 1024; jc += 16) {      // 64 K-chunks of 16
      // A-fragments for this chunk (shared by the 4 N-tiles).
      v2f A[4];
      #pragma unroll
      for (int kk = 0; kk < 4; ++kk) {
        const int j = jc + kk * 4;
        float hi = lutHi[j >> 5];
        v2f lo   = *(const v2f*)&lutLo[(j & 31) + koff];
        A[kk].x = hi * lo.x;
        A[kk].y = hi * lo.y;
      }
      const float* pj = wp + (size_t)jc * 2048;   // uniform base (SGPRs)
      #pragma unroll
      for (int nt = 0; nt < 4; ++nt) {
        #pragma unroll
        for (int kk = 0; kk < 4; ++kk) {
          v4f b = *(const v4f*)(pj + laneOfs + kk * 8192 + nt * 64);
          v2f bre = { b.x, b.y };
          v2f bim = { b.z, b.w };
          accR[nt] = __builtin_amdgcn_wmma_f32_16x16x4_f32(false, A[kk], false, bre,
                                                           (short)0, accR[nt], false, false);
          accI[nt] = __builtin_amdgcn_wmma_f32_16x16x4_f32(false, A[kk], false, bim,
                                                           (short)0, accI[nt], false, false);
        }
      }
    }

    // Epilogue: |Phi|^2 against Z sign masks -> LDS atomic accumulation.
    #pragma unroll
    for (int nt = 0; nt < 4; ++nt) {
      const int nidx = (tbase + nt) * 16 + row;         // global state index N
      #pragma unroll
      for (int r = 0; r < 8; ++r) {
        const int m = r + khalf * 8;                    // local batch row
        const float p = accR[nt][r] * accR[nt][r] + accI[nt][r] * accI[nt][r];
        #pragma unroll
        for (int wq = 0; wq < 10; ++wq) {
          float v = ((nidx >> (9 - wq)) & 1) ? -p : p;
          atomicAdd(&oacc[m * 10 + wq], v);
        }
      }
    }
  }
  __syncthreads();
  for (int i = tid; i < 160; i += 256) out[(size_t)mbase * 10 + i] = oacc[i];
}

// ---------------------------------------------------------------------------
extern "C" void kernel_launch(void* const* d_in, const int* in_sizes, int n_in,
                              void* d_out, int out_size, void* d_ws, size_t ws_size,
                              hipStream_t stream) {
  const float* x       = (const float*)d_in[0];   // (16384, 10)
  const float* weights = (const float*)d_in[1];   // (3, 10, 3)
  float* out = (float*)d_out;                     // (16384, 10)
  float* ws  = (float*)d_ws;

  float* wp   = ws;                 // 2M floats (Re/Im interleaved packed W^T)
  float* lutb = ws + (2u << 20);    // 1M floats

  build_operator_kernel<<<1024, 32, 0, stream>>>(weights, wp);
  encode_lut_kernel<<<64, 256, 0, stream>>>(x, lutb);
  gemm_expect_kernel<<<1024, 256, 0, stream>>>(wp, lutb, out);
}